// KnowledgeGuidedTextGenerator_11338713661450
// MI455X (gfx1250) — compile-verified
//
#include <hip/hip_runtime.h>
#include <hip/hip_bf16.h>
#include <math.h>

typedef float v2f __attribute__((ext_vector_type(2)));
typedef float v8f __attribute__((ext_vector_type(8)));

#define B_DIM 64
#define V_DIM 8192
#define DM    256
#define ED    128
#define NREL  4

// ---------------------------------------------------------------------------
// Kernel 1: entity_emb[64,128] = hidden[64,256] @ W[256,128] + bias[128]
// One wave per 16x16 output tile: 4 m-tiles * 8 n-tiles = 32 waves = 4 blocks.
// ---------------------------------------------------------------------------
__global__ __launch_bounds__(256) void k_to_entity(const float* __restrict__ hidden,
                                                   const float* __restrict__ W,
                                                   const float* __restrict__ bias,
                                                   float* __restrict__ emb) {
    const int lane  = threadIdx.x & 31;
    const int wave  = threadIdx.x >> 5;
    const int t     = blockIdx.x * 8 + wave;     // 0..31
    const int mtile = t >> 3;                    // 0..3
    const int ntile = t & 7;                     // 0..7
    const int mrow  = mtile * 16 + (lane & 15);  // A-fragment row
    const int ncol  = ntile * 16 + (lane & 15);  // B-fragment col
    const int koff  = (lane >> 4) * 2;           // K offset within 4-chunk

    v8f acc = {};
    #pragma unroll 8
    for (int ks = 0; ks < DM; ks += 4) {
        const int k = ks + koff;
        v2f a = { hidden[mrow * DM + k], hidden[mrow * DM + k + 1] };
        v2f b = { W[k * ED + ncol],      W[(k + 1) * ED + ncol] };
        acc = __builtin_amdgcn_wmma_f32_16x16x4_f32(false, a, false, b,
                                                    (short)0, acc, false, false);
    }
    const float bv    = bias[ncol];
    const int   mbase = mtile * 16 + (lane >> 4) * 8;
    #pragma unroll
    for (int r = 0; r < 8; ++r)
        emb[(mbase + r) * ED + ncol] = acc[r] + bv;
}

// ---------------------------------------------------------------------------
// Kernel 2: sims[64,8192] = emb[64,128] @ E[8192,128]^T
// One wave per 16x16 tile: 4 m-tiles * 512 n-tiles = 2048 waves = 256 blocks.
// Consecutive waves share the same n-tile (B rows) across the 4 m-tiles.
// Both operands are K-contiguous -> clean global_load_b64 fragments.
// ---------------------------------------------------------------------------
__global__ __launch_bounds__(256) void k_sims(const float* __restrict__ emb,
                                              const float* __restrict__ E,
                                              float* __restrict__ sims) {
    const int lane  = threadIdx.x & 31;
    const int wave  = threadIdx.x >> 5;
    const int t     = blockIdx.x * 8 + wave;     // 0..2047
    const int mtile = t & 3;                     // 0..3
    const int ntile = t >> 2;                    // 0..511
    const int mrow  = mtile * 16 + (lane & 15);
    const int ncol  = ntile * 16 + (lane & 15);
    const int khalf = (lane >> 4);               // v2f-index offset

    const v2f* __restrict__ pa = (const v2f*)(emb + (size_t)mrow * ED);
    const v2f* __restrict__ pb = (const v2f*)(E + (size_t)ncol * ED);

    v8f acc = {};
    #pragma unroll 8
    for (int ks = 0; ks < ED / 2; ks += 2) {     // each step consumes K=4
        v2f a = pa[ks + khalf];
        v2f b = pb[ks + khalf];
        acc = __builtin_amdgcn_wmma_f32_16x16x4_f32(false, a, false, b,
                                                    (short)0, acc, false, false);
    }
    const int mbase = mtile * 16 + (lane >> 4) * 8;
    #pragma unroll
    for (int r = 0; r < 8; ++r)
        sims[(size_t)(mbase + r) * V_DIM + ncol] = acc[r];
}

// ---------------------------------------------------------------------------
// Kernel 3: closest[b] = argmax_n sims[b, n]   (first occurrence on ties)
// One block (256 threads) per batch row; LDS tree reduction.
// ---------------------------------------------------------------------------
__global__ __launch_bounds__(256) void k_argmax(const float* __restrict__ sims,
                                                int* __restrict__ closest) {
    __shared__ float sv[256];
    __shared__ int   si[256];
    const int b = blockIdx.x;
    const float* row = sims + (size_t)b * V_DIM;

    float best = -INFINITY;
    int   bi   = 0;
    for (int j = threadIdx.x; j < V_DIM; j += 256) {  // j increases -> first-hit tie rule
        float v = row[j];
        if (v > best) { best = v; bi = j; }
    }
    sv[threadIdx.x] = best;
    si[threadIdx.x] = bi;
    __syncthreads();
    for (int s = 128; s > 0; s >>= 1) {
        if (threadIdx.x < s) {
            float ov = sv[threadIdx.x + s];
            int   oi = si[threadIdx.x + s];
            if (ov > sv[threadIdx.x] ||
                (ov == sv[threadIdx.x] && oi < si[threadIdx.x])) {
                sv[threadIdx.x] = ov;
                si[threadIdx.x] = oi;
            }
        }
        __syncthreads();
    }
    if (threadIdx.x == 0) closest[b] = si[0];
}

// ---------------------------------------------------------------------------
// Kernel 4: gather relation rows for closest[b], OR over relations (>0.5),
// empty-set fallback, mask logits with -inf. One block per batch row.
// ---------------------------------------------------------------------------
__global__ __launch_bounds__(256) void k_mask(const float* __restrict__ logits,
                                              const float* __restrict__ relations,
                                              const int* __restrict__ closest,
                                              float* __restrict__ out) {
    __shared__ int s_any;
    const int b = blockIdx.x;
    const int c = closest[b];
    const size_t VV = (size_t)V_DIM * V_DIM;
    const float* __restrict__ rbase = relations + (size_t)c * V_DIM;

    unsigned mask = 0u;
    #pragma unroll 1
    for (int k = 0; k < V_DIM / 256; ++k) {
        const int j = threadIdx.x + k * 256;
        bool valid = false;
        #pragma unroll
        for (int r = 0; r < NREL; ++r)
            valid |= rbase[(size_t)r * VV + j] > 0.5f;
        if (valid) mask |= (1u << k);
    }

    if (threadIdx.x == 0) s_any = 0;
    __syncthreads();
    if (mask != 0u) s_any = 1;                 // benign race: all writers store 1
    __syncthreads();
    const bool rowAny = (s_any != 0);

    const float* __restrict__ lrow = logits + (size_t)b * V_DIM;
    float* __restrict__ orow       = out + (size_t)b * V_DIM;
    #pragma unroll 1
    for (int k = 0; k < V_DIM / 256; ++k) {
        const int j = threadIdx.x + k * 256;
        const bool valid = (mask >> k) & 1u;
        orow[j] = (valid || !rowAny) ? lrow[j] : -INFINITY;
    }
}

// ---------------------------------------------------------------------------
// Launch
// ---------------------------------------------------------------------------
extern "C" void kernel_launch(void* const* d_in, const int* in_sizes, int n_in,
                              void* d_out, int out_size, void* d_ws, size_t ws_size,
                              hipStream_t stream) {
    const float* logits    = (const float*)d_in[0];  // [64, 8192]
    const float* hidden    = (const float*)d_in[1];  // [64, 256]
    const float* W         = (const float*)d_in[2];  // [256, 128]
    const float* bias      = (const float*)d_in[3];  // [128]
    const float* E         = (const float*)d_in[4];  // [8192, 128]
    const float* relations = (const float*)d_in[5];  // [4, 8192, 8192]
    float* out = (float*)d_out;                      // [64, 8192]

    float* emb     = (float*)d_ws;                   // 64*128 floats
    float* sims    = emb + (size_t)B_DIM * ED;       // 64*8192 floats
    int*   closest = (int*)(sims + (size_t)B_DIM * V_DIM);  // 64 ints

    k_to_entity<<<4,   256, 0, stream>>>(hidden, W, bias, emb);
    k_sims     <<<256, 256, 0, stream>>>(emb, E, sims);
    k_argmax   <<<64,  256, 0, stream>>>(sims, closest);
    k_mask     <<<64,  256, 0, stream>>>(logits, relations, closest, out);
}